// SparseGCNConv_58411555225958
// MI455X (gfx1250) — compile-verified
//
#include <hip/hip_runtime.h>
#include <hip/hip_bf16.h>

// SparseGCNConv for MI455X (gfx1250, wave32).
//
// Rewrite: out = segment_sum(norm * (features @ W^T)[src], dst) + bias
// Phase 1: Y = features @ W^T via V_WMMA_F32_16X16X4_F32 (f32-exact, GEMM is
//          only 3.3 GFLOP -- negligible vs the edge phase).
// Phase 2: out[n,:] = bias  (fused bias, nodes with no in-edges get bias).
// Phase 3: out[dst[e],:] += norm[e] * Y[src[e],:]  (global_atomic_add_f32;
//          Y/out are 51.2 MB each -> resident in the 192 MB L2, so this is
//          L2-atomic bound, not HBM bound).

typedef __attribute__((ext_vector_type(2))) float v2f;
typedef __attribute__((ext_vector_type(8))) float v8f;

#define GCN_D 128  // D_IN == D_OUT == 128

// ---------------------------------------------------------------------------
// Kernel 1: Y[n, o] = sum_k features[n, k] * weight[o, k]
// One wave per 16x16 tile of Y. blockDim = (32, 8): 8 waves = 8 N-tiles
// (covers all 128 output columns), blockIdx.x = M-tile (16 nodes).
//
// f32 WMMA 16x16x4 VGPR layouts (ISA 7.12.2):
//   A (16x4, 2 VGPRs): lane l -> row m = l&15; lanes 0-15 hold K={0,1},
//                      lanes 16-31 hold K={2,3}  -> contiguous float2 in K.
//   B (4x16, 2 VGPRs): lane l -> col n = l&15; lanes 0-15 hold K={0,1},
//                      lanes 16-31 hold K={2,3}. B[k][o] = W[o][k], so the
//                      per-lane fragment is a contiguous float2 of row W[o].
//   C/D (16x16, 8 VGPRs): lane l -> col n = l&15, VGPR r -> row r + 8*(l>>4).
// ---------------------------------------------------------------------------
__global__ __launch_bounds__(256) void gcn_gemm_xw(
    const float* __restrict__ A,   // [n_nodes, 128] features
    const float* __restrict__ W,   // [128, 128] weight (row o = output feat)
    float* __restrict__ Y,         // [n_nodes, 128] out
    int n_nodes) {
  const int lane = threadIdx.x;        // 0..31 within wave
  const int row0 = blockIdx.x * 16;    // M-tile base (node rows)
  const int col0 = threadIdx.y * 16;   // N-tile base (output features)
  if (row0 >= n_nodes) return;         // uniform per block; never taken here

  const int mn = lane & 15;            // A row / B col within tile
  const int ks = (lane >> 4) * 2;      // K sub-offset {0, 2}

  const float* arow = A + (size_t)(row0 + mn) * GCN_D + ks;
  const float* wrow = W + (size_t)(col0 + mn) * GCN_D + ks;

  v8f c = {};
#pragma unroll
  for (int k = 0; k < GCN_D; k += 4) {
    v2f a = *(const v2f*)(arow + k);   // 8B aligned (k%4==0, ks in {0,2})
    v2f b = *(const v2f*)(wrow + k);
    // D = A x B + C  (8 args: neg_a, A, neg_b, B, c_mod, C, reuse_a, reuse_b)
    c = __builtin_amdgcn_wmma_f32_16x16x4_f32(
        /*neg_a=*/false, a, /*neg_b=*/false, b,
        /*c_mod=*/(short)0, c, /*reuse_a=*/false, /*reuse_b=*/false);
  }

  const int n  = lane & 15;
  const int mo = (lane >> 4) * 8;
#pragma unroll
  for (int r = 0; r < 8; ++r) {
    Y[(size_t)(row0 + mo + r) * GCN_D + col0 + n] = c[r];
  }
}

// ---------------------------------------------------------------------------
// Kernel 2: out[i] = bias[i & 127]   (bias broadcast; also the identity for
// nodes that receive no messages)
// ---------------------------------------------------------------------------
__global__ __launch_bounds__(256) void gcn_bias_init(
    float* __restrict__ out, const float* __restrict__ bias, size_t total) {
  size_t i = (size_t)blockIdx.x * blockDim.x + threadIdx.x;
  if (i < total) out[i] = bias[i & (GCN_D - 1)];
}

// ---------------------------------------------------------------------------
// Kernel 3: per-edge gather + scaled scatter-add.
// One wave32 per edge per iteration; lane covers float4 of the 128-wide row.
// Grid-stride x4 with a prefetch of the next edge's source row to overlap
// the random L2 gather with the atomic stream.
// ---------------------------------------------------------------------------
__global__ __launch_bounds__(256) void gcn_edge_scatter(
    const float* __restrict__ Y,      // [n_nodes, 128]
    const float* __restrict__ norm,   // [n_edges]
    const int* __restrict__ src,      // [n_edges]
    const int* __restrict__ dst,      // [n_edges]
    float* __restrict__ out,          // [n_nodes, 128]
    int n_edges) {
  const int lane   = threadIdx.x & 31;
  const int wave   = (int)((blockIdx.x * (size_t)blockDim.x + threadIdx.x) >> 5);
  const int nwaves = (int)(((size_t)gridDim.x * blockDim.x) >> 5);

  for (int e = wave; e < n_edges; e += nwaves) {
    const int en = e + nwaves;
    if (en < n_edges) {
      // Pull the next source row toward this WGP (global_prefetch_b8).
      __builtin_prefetch(Y + (size_t)src[en] * GCN_D + lane * 4, 0, 0);
    }
    const float w = norm[e];
    const float4 v = *(const float4*)(Y + (size_t)src[e] * GCN_D + lane * 4);
    float* orow = out + (size_t)dst[e] * GCN_D + lane * 4;
    atomicAdd(orow + 0, v.x * w);   // global_atomic_add_f32 -> L2
    atomicAdd(orow + 1, v.y * w);
    atomicAdd(orow + 2, v.z * w);
    atomicAdd(orow + 3, v.w * w);
  }
}

// ---------------------------------------------------------------------------
extern "C" void kernel_launch(void* const* d_in, const int* in_sizes, int n_in,
                              void* d_out, int out_size, void* d_ws, size_t ws_size,
                              hipStream_t stream) {
  const float* features = (const float*)d_in[0];  // [N, 128]
  const float* norm     = (const float*)d_in[1];  // [E, 1]
  const int*   src      = (const int*)d_in[2];    // [E]
  const int*   dst      = (const int*)d_in[3];    // [E]
  const float* weight   = (const float*)d_in[4];  // [128, 128]
  const float* bias     = (const float*)d_in[5];  // [128]
  float*       out      = (float*)d_out;          // [N, 128]

  const int n_nodes = in_sizes[0] / GCN_D;
  const int n_edges = in_sizes[2];

  float* Y = (float*)d_ws;  // needs n_nodes * 128 * 4 B = 51.2 MB of scratch

  // Phase 1: Y = features @ W^T  (WMMA f32)
  dim3 gblock(32, 8);
  gcn_gemm_xw<<<(n_nodes + 15) / 16, gblock, 0, stream>>>(features, weight, Y,
                                                          n_nodes);

  // Phase 2: out = bias (broadcast per row)
  const size_t total = (size_t)n_nodes * GCN_D;
  gcn_bias_init<<<(int)((total + 255) / 256), 256, 0, stream>>>(out, bias,
                                                                total);

  // Phase 3: scatter-add messages. ~4 edges per wave.
  const int waves  = (n_edges + 3) / 4;
  const int blocks = (waves + 7) / 8;  // 8 waves per 256-thread block
  gcn_edge_scatter<<<blocks, 256, 0, stream>>>(Y, norm, src, dst, out,
                                               n_edges);
}